// perm_inv_n_63651415327410
// MI455X (gfx1250) — compile-verified
//
#include <hip/hip_runtime.h>

typedef __attribute__((ext_vector_type(2))) float v2f;
typedef __attribute__((ext_vector_type(4))) float v4f;
typedef __attribute__((ext_vector_type(8))) float v8f;

#define Z_DIM   256
#define A_DIM   512
#define B_DIM   64
#define STRIDE  266432   // per-z output floats: 64 + 4096 + 64 + 64 + 262144
#define OFF_H1  0
#define OFF_H2  64
#define OFF_H3  4160
#define OFF_H4  4224
#define OFF_H5  4288

#define ROWS    16                    // a-rows staged per pipeline stage
#define NSTAGES (A_DIM / ROWS)        // 32

// ---------------------------------------------------------------------------
// Kernel 1: per-z gram (WMMA f32 16x16x4), column sums, column sum-of-squares.
// 1 block per z, 128 threads = 4 waves. Wave w computes gram rows [16w,16w+16).
// Software-pipelined: double-buffered LDS slabs, register prefetch of stage
// st+1 issued before the barrier so global latency overlaps WMMA work.
// ---------------------------------------------------------------------------
__global__ __launch_bounds__(128) void gram_cs_kernel(const float* __restrict__ x,
                                                      float* __restrict__ out) {
  const int z    = blockIdx.x;
  const int tid  = threadIdx.x;
  const int wave = tid >> 5;
  const int lane = tid & 31;
  const int l15  = lane & 15;
  const int hi   = lane >> 4;   // 0: lanes 0-15, 1: lanes 16-31
  const int kb   = hi * 2;      // K sub-base held by this lane (A/B frag layout)

  __shared__ float slab[2][ROWS * B_DIM];   // 2 x 4KB ping-pong

  v8f acc0 = {}, acc1 = {}, acc2 = {}, acc3 = {};
  float cs = 0.0f, sq = 0.0f;

  const float* xz = x + (size_t)z * (A_DIM * B_DIM);

  // Prefetch stage 0 into registers (8 floats / thread = 1024 floats / block).
  v4f r0 = ((const v4f*)xz)[tid * 2 + 0];
  v4f r1 = ((const v4f*)xz)[tid * 2 + 1];

  for (int st = 0; st < NSTAGES; ++st) {
    float* s = slab[st & 1];
    ((v4f*)s)[tid * 2 + 0] = r0;
    ((v4f*)s)[tid * 2 + 1] = r1;

    if (st + 1 < NSTAGES) {  // prefetch next stage; latency hidden by compute
      const v4f* nsrc = (const v4f*)(xz + (st + 1) * ROWS * B_DIM);
      r0 = nsrc[tid * 2 + 0];
      r1 = nsrc[tid * 2 + 1];
    }

    __syncthreads();  // slab[st&1] visible; also fences reads of slab[(st+1)&1]

    // cs / sum-of-squares partials (threads 0..63 cover the b axis).
    if (tid < B_DIM) {
      #pragma unroll
      for (int r = 0; r < ROWS; ++r) {
        float v = s[r * B_DIM + tid];
        cs += v;
        sq += v * v;
      }
    }

    // 4 K-steps of 4 per stage; A fragment reused across the 4 column tiles.
    #pragma unroll
    for (int kk = 0; kk < 4; ++kk) {
      const float* sk0 = s + (kk * 4 + kb + 0) * B_DIM;
      const float* sk1 = s + (kk * 4 + kb + 1) * B_DIM;

      v2f a;
      a.x = sk0[wave * 16 + l15];
      a.y = sk1[wave * 16 + l15];

      v2f b0, b1, b2, b3;
      b0.x = sk0[ 0 + l15];  b0.y = sk1[ 0 + l15];
      b1.x = sk0[16 + l15];  b1.y = sk1[16 + l15];
      b2.x = sk0[32 + l15];  b2.y = sk1[32 + l15];
      b3.x = sk0[48 + l15];  b3.y = sk1[48 + l15];

      acc0 = __builtin_amdgcn_wmma_f32_16x16x4_f32(false, a, false, b0, (short)0, acc0, false, false);
      acc1 = __builtin_amdgcn_wmma_f32_16x16x4_f32(false, a, false, b1, (short)0, acc1, false, false);
      acc2 = __builtin_amdgcn_wmma_f32_16x16x4_f32(false, a, false, b2, (short)0, acc2, false, false);
      acc3 = __builtin_amdgcn_wmma_f32_16x16x4_f32(false, a, false, b3, (short)0, acc3, false, false);
    }
  }

  float* oz = out + (size_t)z * STRIDE;

  // D-matrix layout (ISA 7.12.2): VGPR j -> M=j (lanes 0-15) / M=j+8 (lanes 16-31), N=lane&15.
  #pragma unroll
  for (int j = 0; j < 8; ++j) {
    const int row = wave * 16 + j + hi * 8;
    float* gr = oz + OFF_H2 + row * B_DIM + l15;
    gr[ 0] = acc0[j];
    gr[16] = acc1[j];
    gr[32] = acc2[j];
    gr[48] = acc3[j];
  }

  if (tid < B_DIM) {
    oz[OFF_H1 + tid] = cs;        // h1
    oz[OFF_H3 + tid] = sq;        // h3
    oz[OFF_H4 + tid] = cs * cs;   // h4
  }
}

// ---------------------------------------------------------------------------
// Kernel 2: h5[z,b,c,d] = gram[z,b,c] * cs[z,d]. Pure streaming write (256 MB);
// two b128 non-temporal stores per thread (output exceeds the 192MB L2, so NT
// keeps gram/cs lines resident). Reads come from regions kernel 1 wrote
// into d_out (L2-hot). Stores are fully coalesced.
// ---------------------------------------------------------------------------
__global__ __launch_bounds__(256) void h5_kernel(float* out) {
  const size_t t  = (size_t)blockIdx.x * 256 + threadIdx.x;
  const size_t i4 = t * 2;               // first v4f index in h5 space
  const int z  = (int)(i4 >> 16);        // 65536 v4f per z
  const int r  = (int)(i4 & 65535);      // v4f within this z's h5 (even)
  const int bc = r >> 4;                 // 0..4095  (b*64 + c); same for r and r+1
  const int d4 = r & 15;                 // which v4f of cs

  float* oz = out + (size_t)z * STRIDE;
  const float g = oz[OFF_H2 + bc];
  const v4f  c0 = ((const v4f*)(oz + OFF_H1))[d4 + 0];
  const v4f  c1 = ((const v4f*)(oz + OFF_H1))[d4 + 1];

  v4f* dst = (v4f*)(oz + OFF_H5) + r;
  __builtin_nontemporal_store(g * c0, dst + 0);
  __builtin_nontemporal_store(g * c1, dst + 1);
}

extern "C" void kernel_launch(void* const* d_in, const int* in_sizes, int n_in,
                              void* d_out, int out_size, void* d_ws, size_t ws_size,
                              hipStream_t stream) {
  const float* x = (const float*)d_in[0];
  float* out = (float*)d_out;

  // Kernel 1: gram + cs + sq + cs^2  (256 blocks, 4 waves each, pipelined)
  gram_cs_kernel<<<Z_DIM, 128, 0, stream>>>(x, out);

  // Kernel 2: h5 outer-product stream. 16,777,216 v4f / 2 per thread.
  h5_kernel<<<32768, 256, 0, stream>>>(out);
}